// DualFusionLayer_27642409517764
// MI455X (gfx1250) — compile-verified
//
#include <hip/hip_runtime.h>
#include <math.h>

// Problem constants (match reference)
#define Bc 8
#define Lc 512
#define Dc 768
#define Hc 12
#define HDc 64
#define Ac 3
#define INNERc 3072
#define BLc (Bc*Lc)                 // 4096 rows
#define BLDc ((size_t)Bc*Lc*Dc)     // 3,145,728 elements

typedef float v2f __attribute__((ext_vector_type(2)));
typedef float v8f __attribute__((ext_vector_type(8)));

// Tiling: block = 256 threads = 8 wave32, arranged 4x2; each wave owns a
// 16x32 C tile (two 16x16 accumulators sharing one A fragment per K-step).
// Block tile: 64x64, K staged through LDS in chunks of 64.
// Both LDS tiles are K-contiguous per row so fragments load as single b64s.
#define BM 64
#define BN 64
#define KC 64
#define KP (KC + 2)                 // pad: frag reads hit distinct banks

// ---------------------------------------------------------------------------
// General tiled fp32 WMMA GEMM:
//   C[z] (+)= act( A[z][M,K] * op(B[z])[K,N] + bias[z] )
// Batch z decomposed as zb=z/Hdiv, zh=z%Hdiv with per-operand (b,h) strides,
// so the same kernel does plain GEMMs, per-(b,h) QK^T (transB), and P*V with
// head-sliced ld=768 views.
// ---------------------------------------------------------------------------
__global__ __launch_bounds__(256)
void gemm_wmma_f32(const float* __restrict__ Ag, const float* __restrict__ Bg,
                   const float* __restrict__ bias, float* __restrict__ Cg,
                   int M, int N, int K, int lda, int ldb, int ldc,
                   long long sAb, long long sAh, long long sBb, long long sBh,
                   long long sCb, long long sCh, int Hdiv, long long sBiasZ,
                   int transB, int accumulate, int doGelu)
{
    __shared__ float As [BM][KP];   // As[m][k]
    __shared__ float BsT[BN][KP];   // BsT[n][k]  (column-major staging of B)

    const int z  = blockIdx.z;
    const int zb = z / Hdiv;
    const int zh = z % Hdiv;
    const float* A  = Ag + zb * sAb + zh * sAh;
    const float* Bm = Bg + zb * sBb + zh * sBh;
    float*       C  = Cg + zb * sCb + zh * sCh;

    const int tid  = threadIdx.x;
    const int lane = tid & 31;
    const int wave = tid >> 5;
    const int wm = (wave >> 1) << 4;   // 0,16,32,48 (wave row)
    const int wn = (wave & 1) << 5;    // 0,32       (wave col, 32 wide)
    const int r  = lane & 15;          // row (A) / col (B) within 16x16 frag
    const int kh = lane >> 4;          // K half-select within 16x16x4 frag

    const int rowBase = blockIdx.y * BM;
    const int colBase = blockIdx.x * BN;

    v8f acc0 = {0.f, 0.f, 0.f, 0.f, 0.f, 0.f, 0.f, 0.f};
    v8f acc1 = {0.f, 0.f, 0.f, 0.f, 0.f, 0.f, 0.f, 0.f};

    // Staging: thread -> (source row 0..63, 16 consecutive cols) for both operands
    const int sr = tid >> 2;           // 0..63
    const int sc = (tid & 3) << 4;     // 0,16,32,48

    for (int kb = 0; kb < K; kb += KC) {
        {
            const float* ap = A + (size_t)(rowBase + sr) * lda + kb + sc;
            #pragma unroll
            for (int j = 0; j < 16; ++j) As[sr][sc + j] = ap[j];
        }
        if (!transB) {
            // B row-major [K,N]: read K-row sr, 16 consecutive cols; scatter
            // into column-major BsT.
            const float* bp = Bm + (size_t)(kb + sr) * ldb + colBase + sc;
            #pragma unroll
            for (int j = 0; j < 16; ++j) BsT[sc + j][sr] = bp[j];
        } else {
            // Bt row-major [N,K]: read N-row sr, 16 consecutive k; contiguous.
            const float* tp = Bm + (size_t)(colBase + sr) * ldb + kb + sc;
            #pragma unroll
            for (int j = 0; j < 16; ++j) BsT[sr][sc + j] = tp[j];
        }
        __syncthreads();

        #pragma unroll
        for (int kk = 0; kk < KC; kk += 4) {
            v2f a, b0, b1;
            // 16x4 fp32 A frag: lanes0-15 -> K=kk,kk+1 ; lanes16-31 -> kk+2,kk+3
            a.x  = As[wm + r][kk + 2 * kh];
            a.y  = As[wm + r][kk + 2 * kh + 1];
            // two 4x16 fp32 B frags sharing the A frag (contiguous b64 loads)
            b0.x = BsT[wn + r][kk + 2 * kh];
            b0.y = BsT[wn + r][kk + 2 * kh + 1];
            b1.x = BsT[wn + 16 + r][kk + 2 * kh];
            b1.y = BsT[wn + 16 + r][kk + 2 * kh + 1];
            acc0 = __builtin_amdgcn_wmma_f32_16x16x4_f32(
                false, a, false, b0, (short)0, acc0, false, false);
            acc1 = __builtin_amdgcn_wmma_f32_16x16x4_f32(
                false, a, false, b1, (short)0, acc1, false, false);
        }
        __syncthreads();
    }

    // Epilogue: D layout -> VGPR v holds row (wm + v + 8*kh), col (wn + r)
    const int col0 = colBase + wn + r;
    const int col1 = col0 + 16;
    const float* bp = bias ? (bias + zb * sBiasZ) : nullptr;
    const float bv0 = bp ? bp[col0] : 0.f;
    const float bv1 = bp ? bp[col1] : 0.f;
    #pragma unroll
    for (int v = 0; v < 8; ++v) {
        const int row = rowBase + wm + v + 8 * kh;
        float val0 = acc0[v] + bv0;
        float val1 = acc1[v] + bv1;
        if (doGelu) {
            val0 = 0.5f * val0 * (1.f + erff(val0 * 0.70710678118654752f));
            val1 = 0.5f * val1 * (1.f + erff(val1 * 0.70710678118654752f));
        }
        const size_t i0 = (size_t)row * ldc + col0;
        const size_t i1 = (size_t)row * ldc + col1;
        if (accumulate) { C[i0] += val0; C[i1] += val1; }
        else            { C[i0]  = val0; C[i1]  = val1; }
    }
}

// ---------------------------------------------------------------------------
// In-place row softmax over L=512 with fused scale + mask coefficient:
//   x_j = S_j*scale + mcoef*mask[b,i,j] ; S_j = softmax(x)_j
// grid = (L, B*H), block = 256 (2 elems/thread)
// ---------------------------------------------------------------------------
__global__ __launch_bounds__(256)
void softmax_mask(float* __restrict__ S, const float* __restrict__ mask,
                  float scale, float mcoef)
{
    const int i = blockIdx.x;
    const int z = blockIdx.y;          // b*H + h
    const int b = z / Hc;
    float* row = S + ((size_t)z * Lc + i) * Lc;
    const float* mrow = mask + ((size_t)b * Lc + i) * Lc;

    const int t = threadIdx.x;
    float x0 = row[t] * scale + mcoef * mrow[t];
    float x1 = row[t + 256] * scale + mcoef * mrow[t + 256];

    __shared__ float red[256];
    red[t] = fmaxf(x0, x1);
    __syncthreads();
    for (int s = 128; s > 0; s >>= 1) {
        if (t < s) red[t] = fmaxf(red[t], red[t + s]);
        __syncthreads();
    }
    const float m = red[0];
    __syncthreads();

    const float e0 = expf(x0 - m);
    const float e1 = expf(x1 - m);
    red[t] = e0 + e1;
    __syncthreads();
    for (int s = 128; s > 0; s >>= 1) {
        if (t < s) red[t] += red[t + s];
        __syncthreads();
    }
    const float inv = 1.f / red[0];
    row[t]       = e0 * inv;
    row[t + 256] = e1 * inv;
}

// ---------------------------------------------------------------------------
// y = LayerNorm(x + r) * g + be, rows of length D=768 (3 elems/thread)
// ---------------------------------------------------------------------------
__global__ __launch_bounds__(256)
void add_layernorm(const float* __restrict__ X, const float* __restrict__ R,
                   const float* __restrict__ g, const float* __restrict__ be,
                   float* __restrict__ Y)
{
    const int row = blockIdx.x;
    const float* x  = X + (size_t)row * Dc;
    const float* rr = R + (size_t)row * Dc;
    float* y = Y + (size_t)row * Dc;
    const int t = threadIdx.x;

    float v0 = x[t] + rr[t];
    float v1 = x[t + 256] + rr[t + 256];
    float v2 = x[t + 512] + rr[t + 512];

    __shared__ float red[256];
    red[t] = v0 + v1 + v2;
    __syncthreads();
    for (int s = 128; s > 0; s >>= 1) {
        if (t < s) red[t] += red[t + s];
        __syncthreads();
    }
    const float mean = red[0] * (1.f / Dc);
    __syncthreads();

    const float d0 = v0 - mean, d1 = v1 - mean, d2 = v2 - mean;
    red[t] = d0 * d0 + d1 * d1 + d2 * d2;
    __syncthreads();
    for (int s = 128; s > 0; s >>= 1) {
        if (t < s) red[t] += red[t + s];
        __syncthreads();
    }
    const float inv = rsqrtf(red[0] * (1.f / Dc) + 1e-12f);
    y[t]       = d0 * inv * g[t]       + be[t];
    y[t + 256] = d1 * inv * g[t + 256] + be[t + 256];
    y[t + 512] = d2 * inv * g[t + 512] + be[t + 512];
}

// ---------------------------------------------------------------------------
static inline void launch_gemm(hipStream_t st, const float* A, const float* B,
                               const float* bias, float* C,
                               int M, int N, int K, int lda, int ldb, int ldc,
                               long long sAb, long long sAh, long long sBb,
                               long long sBh, long long sCb, long long sCh,
                               int Hdiv, long long sBiasZ,
                               int transB, int accN, int gelu, int Z)
{
    dim3 grid(N / BN, M / BM, Z);
    gemm_wmma_f32<<<grid, 256, 0, st>>>(A, B, bias, C, M, N, K, lda, ldb, ldc,
                                        sAb, sAh, sBb, sBh, sCb, sCh,
                                        Hdiv, sBiasZ, transB, accN, gelu);
}

extern "C" void kernel_launch(void* const* d_in, const int* in_sizes, int n_in,
                              void* d_out, int out_size, void* d_ws, size_t ws_size,
                              hipStream_t stream)
{
    (void)in_sizes; (void)n_in; (void)out_size; (void)ws_size;

    const float* E_v     = (const float*)d_in[0];
    const float* E_attrs = (const float*)d_in[1];
    const float* E_va    = (const float*)d_in[2];
    const float* mask    = (const float*)d_in[3];
    const float* Wq_id   = (const float*)d_in[4];
    const float* bq_id   = (const float*)d_in[5];
    const float* Wk_id   = (const float*)d_in[6];
    const float* bk_id   = (const float*)d_in[7];
    const float* Wq_attr = (const float*)d_in[8];
    const float* bq_attr = (const float*)d_in[9];
    const float* Wk_attr = (const float*)d_in[10];
    const float* bk_attr = (const float*)d_in[11];
    const float* Wq_f = (const float*)d_in[12];
    const float* bq_f = (const float*)d_in[13];
    const float* Wk_f = (const float*)d_in[14];
    const float* bk_f = (const float*)d_in[15];
    const float* Wv_f = (const float*)d_in[16];
    const float* bv_f = (const float*)d_in[17];
    const float* Wo_f = (const float*)d_in[18];
    const float* bo_f = (const float*)d_in[19];
    const float* g1 = (const float*)d_in[20];
    const float* be1 = (const float*)d_in[21];
    const float* g2 = (const float*)d_in[22];
    const float* be2 = (const float*)d_in[23];
    const float* g3 = (const float*)d_in[24];
    const float* be3 = (const float*)d_in[25];
    const float* g4 = (const float*)d_in[26];
    const float* be4 = (const float*)d_in[27];
    const float* W1_id = (const float*)d_in[28];
    const float* b1_id = (const float*)d_in[29];
    const float* W2_id = (const float*)d_in[30];
    const float* b2_id = (const float*)d_in[31];
    const float* W1_f = (const float*)d_in[32];
    const float* b1_f = (const float*)d_in[33];
    const float* W2_f = (const float*)d_in[34];
    const float* b2_f = (const float*)d_in[35];

    // Workspace arena (floats)
    float* ws = (float*)d_ws;
    const size_t SLL = (size_t)Bc * Hc * Lc * Lc;   // 25,165,824
    float* S   = ws;
    float* Q0  = S  + SLL;
    float* K0  = Q0 + BLDc;
    float* QA  = K0 + BLDc;        // 3*BLD (attr Q)
    float* KA  = QA + 3 * BLDc;    // 3*BLD (attr K)
    float* T0  = KA + 3 * BLDc;
    float* T1  = T0 + BLDc;
    float* T2  = T1 + BLDc;
    float* HID = QA;               // reuse dead QA/KA region: 4096x3072 fits in 6*BLD

    float* out0 = (float*)d_out;          // R_v
    float* out1 = out0 + BLDc;            // R_va

    const long long sQb = (long long)Lc * Dc;        // per-batch stride in [B*L, D]
    const long long sSb = (long long)Hc * Lc * Lc;   // scores per-b
    const long long sSh = (long long)Lc * Lc;        // scores per-h

    // ---------------- Phase 1: ID stream + attribute streams ----------------
    launch_gemm(stream, E_v, Wq_id, bq_id, Q0, BLc, Dc, Dc, Dc, Dc, Dc,
                0, 0, 0, 0, 0, 0, 1, 0, 0, 0, 0, 1);
    launch_gemm(stream, E_v, Wk_id, bk_id, K0, BLc, Dc, Dc, Dc, Dc, Dc,
                0, 0, 0, 0, 0, 0, 1, 0, 0, 0, 0, 1);
    launch_gemm(stream, E_attrs, Wq_attr, bq_attr, QA, BLc, Dc, Dc, Dc, Dc, Dc,
                (long long)BLDc, 0, (long long)Dc * Dc, 0, (long long)BLDc, 0,
                1, Dc, 0, 0, 0, Ac);
    launch_gemm(stream, E_attrs, Wk_attr, bk_attr, KA, BLc, Dc, Dc, Dc, Dc, Dc,
                (long long)BLDc, 0, (long long)Dc * Dc, 0, (long long)BLDc, 0,
                1, Dc, 0, 0, 0, Ac);

    // Fused scores: S[b,h] = sum over 4 streams of Q_s K_s^T (raw dot products)
    launch_gemm(stream, Q0, K0, nullptr, S, Lc, Lc, HDc, Dc, Dc, Lc,
                sQb, HDc, sQb, HDc, sSb, sSh, Hc, 0, 1, 0, 0, Bc * Hc);
    for (int a = 0; a < Ac; ++a)
        launch_gemm(stream, QA + (size_t)a * BLDc, KA + (size_t)a * BLDc, nullptr, S,
                    Lc, Lc, HDc, Dc, Dc, Lc,
                    sQb, HDc, sQb, HDc, sSb, sSh, Hc, 0, 1, 1, 0, Bc * Hc);

    // softmax((sumQK/8 + 5*mask)/8) = softmax(sumQK/64 + (5/8)*mask)
    softmax_mask<<<dim3(Lc, Bc * Hc), 256, 0, stream>>>(S, mask, 1.f / 64.f, 5.f / 8.f);

    // context_v = P @ V_v (V = E_v head slices)
    launch_gemm(stream, S, E_v, nullptr, T0, Lc, HDc, Lc, Lc, Dc, Dc,
                sSb, sSh, sQb, HDc, sQb, HDc, Hc, 0, 0, 0, 0, Bc * Hc);

    add_layernorm<<<BLc, 256, 0, stream>>>(T0, E_v, g1, be1, T1);   // R_v1

    launch_gemm(stream, T1, W1_id, b1_id, HID, BLc, INNERc, Dc, Dc, INNERc, INNERc,
                0, 0, 0, 0, 0, 0, 1, 0, 0, 0, 1, 1);                 // gelu
    launch_gemm(stream, HID, W2_id, b2_id, T2, BLc, Dc, INNERc, INNERc, Dc, Dc,
                0, 0, 0, 0, 0, 0, 1, 0, 0, 0, 0, 1);
    add_layernorm<<<BLc, 256, 0, stream>>>(T2, T1, g2, be2, out0);   // R_v

    // ---------------- Phase 2: fused self-attention on E_va ----------------
    launch_gemm(stream, E_va, Wq_f, bq_f, Q0, BLc, Dc, Dc, Dc, Dc, Dc,
                0, 0, 0, 0, 0, 0, 1, 0, 0, 0, 0, 1);
    launch_gemm(stream, E_va, Wk_f, bk_f, K0, BLc, Dc, Dc, Dc, Dc, Dc,
                0, 0, 0, 0, 0, 0, 1, 0, 0, 0, 0, 1);
    launch_gemm(stream, E_va, Wv_f, bv_f, T0, BLc, Dc, Dc, Dc, Dc, Dc,
                0, 0, 0, 0, 0, 0, 1, 0, 0, 0, 0, 1);

    launch_gemm(stream, Q0, K0, nullptr, S, Lc, Lc, HDc, Dc, Dc, Lc,
                sQb, HDc, sQb, HDc, sSb, sSh, Hc, 0, 1, 0, 0, Bc * Hc);
    softmax_mask<<<dim3(Lc, Bc * Hc), 256, 0, stream>>>(S, mask, 1.f / 8.f, 1.f);

    launch_gemm(stream, S, T0, nullptr, T2, Lc, HDc, Lc, Lc, Dc, Dc,
                sSb, sSh, sQb, HDc, sQb, HDc, Hc, 0, 0, 0, 0, Bc * Hc);

    launch_gemm(stream, T2, Wo_f, bo_f, T1, BLc, Dc, Dc, Dc, Dc, Dc,
                0, 0, 0, 0, 0, 0, 1, 0, 0, 0, 0, 1);
    add_layernorm<<<BLc, 256, 0, stream>>>(T1, E_va, g3, be3, T2);   // R_va1

    launch_gemm(stream, T2, W1_f, b1_f, HID, BLc, INNERc, Dc, Dc, INNERc, INNERc,
                0, 0, 0, 0, 0, 0, 1, 0, 0, 0, 1, 1);                 // gelu
    launch_gemm(stream, HID, W2_f, b2_f, T1, BLc, Dc, INNERc, INNERc, Dc, Dc,
                0, 0, 0, 0, 0, 0, 1, 0, 0, 0, 0, 1);
    add_layernorm<<<BLc, 256, 0, stream>>>(T1, T2, g4, be4, out1);   // R_va
}